// ImbalancedGCN_43456479101292
// MI455X (gfx1250) — compile-verified
//
#include <hip/hip_runtime.h>
#include <hip/hip_bf16.h>

typedef float v2f __attribute__((ext_vector_type(2)));
typedef float v8f __attribute__((ext_vector_type(8)));

#define IN_CH 128
#define HID_CH 128
#define OUT_CH 2

// ---------------- degree / normalization ----------------

__global__ void k_init_deg(float* deg, int n) {
    int i = blockIdx.x * blockDim.x + threadIdx.x;
    if (i < n) deg[i] = 1.0f;  // self-loop contributes 1 to in-degree
}

__global__ void k_deg_scatter(const long long* __restrict__ ei, float* deg, long long E) {
    long long e = (long long)blockIdx.x * blockDim.x + threadIdx.x;
    if (e < E) {
        long long c = ei[E + e];  // col = target
        atomicAdd(&deg[c], 1.0f);
    }
}

__global__ void k_dinv(float* deg, int n) {
    int i = blockIdx.x * blockDim.x + threadIdx.x;
    if (i < n) {
        float d = deg[i];
        deg[i] = (d > 0.0f) ? rsqrtf(d) : 0.0f;   // overwrite with D^-1/2
    }
}

// ---------------- layer-1 dense transform: H = X @ W1 (WMMA f32 16x16x4) ----------------
// One wave computes one 16x16 output tile. K = 128 -> 32 WMMA steps.
// f32 A-matrix (16x4): lane L<16 holds row M=L, K = k+0,k+1 in {a.x,a.y};
//                      lane L>=16 holds row M=L-16, K = k+2,k+3.
// f32 B-matrix (4x16): mirrored with N in place of M.
// f32 C/D (16x16): VGPR v -> (M = v + (lane>=16 ? 8 : 0), N = lane&15).

__global__ void k_gemm1_wmma(const float* __restrict__ X, const float* __restrict__ W,
                             float* __restrict__ H, int n_nodes) {
    int gwave = (int)((blockIdx.x * (long long)blockDim.x + threadIdx.x) >> 5);
    int lane  = threadIdx.x & 31;
    int mTile = gwave >> 3;       // 8 N-tiles (128/16) per M-tile
    int nTile = gwave & 7;
    int m0 = mTile * 16;
    if (m0 >= n_nodes) return;    // wave-uniform exit: EXEC stays all-ones for WMMA

    int hl   = lane & 15;
    int koff = (lane >> 4) * 2;   // 0 or 2
    int mrow = m0 + hl;
    if (mrow >= n_nodes) mrow = n_nodes - 1;  // clamp loads for (impossible here) tail tile
    int ncol = nTile * 16 + hl;

    const float* xrow = X + (long long)mrow * IN_CH;
    v8f acc = {0.f, 0.f, 0.f, 0.f, 0.f, 0.f, 0.f, 0.f};

    for (int k = 0; k < IN_CH; k += 4) {
        v2f a = *(const v2f*)(xrow + k + koff);            // 8B-aligned (k+koff even)
        v2f b;
        b.x = W[(k + koff + 0) * HID_CH + ncol];
        b.y = W[(k + koff + 1) * HID_CH + ncol];
        acc = __builtin_amdgcn_wmma_f32_16x16x4_f32(
            /*neg_a=*/false, a, /*neg_b=*/false, b,
            /*c_mod=*/(short)0, acc, /*reuse_a=*/false, /*reuse_b=*/false);
    }

    int mbase = m0 + ((lane >> 4) ? 8 : 0);
    int nc    = nTile * 16 + hl;
#pragma unroll
    for (int v = 0; v < 8; v++) {
        int m = mbase + v;
        if (m < n_nodes) H[(long long)m * HID_CH + nc] = acc[v];
    }
}

// ---------------- layer-1 propagation ----------------

// agg1[i] = dinv[i]^2 * h[i]  (self-loop term; also initializes the buffer)
__global__ void k_init_agg1(const float* __restrict__ h, const float* __restrict__ dinv,
                            float* __restrict__ agg, long long total) {
    long long idx = (long long)blockIdx.x * blockDim.x + threadIdx.x;
    if (idx < total) {
        float di = dinv[idx >> 7];
        agg[idx] = di * di * h[idx];
    }
}

// one wave per edge: agg1[col] += dinv[row]*dinv[col] * h[row]  (128 channels)
__global__ void k_scatter1(const long long* __restrict__ ei, const float* __restrict__ h,
                           const float* __restrict__ dinv, float* __restrict__ agg, long long E) {
    long long wave = ((long long)blockIdx.x * blockDim.x + threadIdx.x) >> 5;
    int lane = threadIdx.x & 31;
    if (wave >= E) return;
    long long r = ei[wave];
    long long c = ei[E + wave];
    float nrm = dinv[r] * dinv[c];
    const float* hr = h + r * HID_CH;
    float* ac = agg + c * HID_CH;
#pragma unroll
    for (int cc = 0; cc < HID_CH; cc += 32) {
        atomicAdd(&ac[cc + lane], nrm * hr[cc + lane]);
    }
}

// h1 = relu(agg1 + b1), in place
__global__ void k_relu_bias(float* __restrict__ agg, const float* __restrict__ b1, long long total) {
    long long idx = (long long)blockIdx.x * blockDim.x + threadIdx.x;
    if (idx < total) {
        float v = agg[idx] + b1[idx & (HID_CH - 1)];
        agg[idx] = v > 0.0f ? v : 0.0f;
    }
}

// ---------------- layer-2 dense transform: h2 = h1 @ W2 (128 -> 2) ----------------
// one wave per node, shuffle reduction
__global__ void k_gemm2(const float* __restrict__ h1, const float* __restrict__ W2,
                        float* __restrict__ h2, int n) {
    long long wave = ((long long)blockIdx.x * blockDim.x + threadIdx.x) >> 5;
    int lane = threadIdx.x & 31;
    if (wave >= n) return;
    const float* row = h1 + wave * HID_CH;
    float s0 = 0.0f, s1 = 0.0f;
#pragma unroll
    for (int c = 0; c < HID_CH; c += 32) {
        float v = row[c + lane];
        s0 += v * W2[(c + lane) * OUT_CH + 0];
        s1 += v * W2[(c + lane) * OUT_CH + 1];
    }
#pragma unroll
    for (int off = 16; off > 0; off >>= 1) {
        s0 += __shfl_xor(s0, off, 32);
        s1 += __shfl_xor(s1, off, 32);
    }
    if (lane == 0) {
        h2[wave * OUT_CH + 0] = s0;
        h2[wave * OUT_CH + 1] = s1;
    }
}

// out[i] = dinv[i]^2 * h2[i] + b2   (self-loop term + bias; initializes d_out)
__global__ void k_init_out(const float* __restrict__ h2, const float* __restrict__ dinv,
                           const float* __restrict__ b2, float* __restrict__ out, int n) {
    int i = blockIdx.x * blockDim.x + threadIdx.x;
    if (i < n) {
        float di2 = dinv[i] * dinv[i];
        out[i * OUT_CH + 0] = di2 * h2[i * OUT_CH + 0] + b2[0];
        out[i * OUT_CH + 1] = di2 * h2[i * OUT_CH + 1] + b2[1];
    }
}

// one thread per edge: out[col] += dinv[row]*dinv[col] * h2[row]  (2 channels)
__global__ void k_scatter2(const long long* __restrict__ ei, const float* __restrict__ h2,
                           const float* __restrict__ dinv, float* __restrict__ out, long long E) {
    long long e = (long long)blockIdx.x * blockDim.x + threadIdx.x;
    if (e >= E) return;
    long long r = ei[e];
    long long c = ei[E + e];
    float nrm = dinv[r] * dinv[c];
    atomicAdd(&out[c * OUT_CH + 0], nrm * h2[r * OUT_CH + 0]);
    atomicAdd(&out[c * OUT_CH + 1], nrm * h2[r * OUT_CH + 1]);
}

// ---------------- launch ----------------

extern "C" void kernel_launch(void* const* d_in, const int* in_sizes, int n_in,
                              void* d_out, int out_size, void* d_ws, size_t ws_size,
                              hipStream_t stream) {
    const float*     x  = (const float*)d_in[0];
    const long long* ei = (const long long*)d_in[1];
    const float*     W1 = (const float*)d_in[2];
    const float*     b1 = (const float*)d_in[3];
    const float*     W2 = (const float*)d_in[4];
    const float*     b2 = (const float*)d_in[5];
    float* out = (float*)d_out;

    const int       N = in_sizes[0] / IN_CH;     // 100000
    const long long E = (long long)in_sizes[1] / 2;  // 3200000

    // workspace layout
    float* deg  = (float*)d_ws;                        // N  (becomes dinv in place)
    float* h    = deg + N;                             // N*128
    float* agg1 = h + (long long)N * HID_CH;           // N*128 (becomes h1)
    float* h2   = agg1 + (long long)N * HID_CH;        // N*2

    const int B = 256;
    long long totalH = (long long)N * HID_CH;

    // normalization
    k_init_deg<<<(N + B - 1) / B, B, 0, stream>>>(deg, N);
    k_deg_scatter<<<(int)((E + B - 1) / B), B, 0, stream>>>(ei, deg, E);
    k_dinv<<<(N + B - 1) / B, B, 0, stream>>>(deg, N);

    // layer 1
    {
        long long mTiles = (N + 15) / 16;
        long long waves  = mTiles * (HID_CH / 16);
        long long thr    = waves * 32;
        k_gemm1_wmma<<<(int)((thr + B - 1) / B), B, 0, stream>>>(x, W1, h, N);
    }
    k_init_agg1<<<(int)((totalH + B - 1) / B), B, 0, stream>>>(h, deg, agg1, totalH);
    k_scatter1<<<(int)((E * 32 + B - 1) / B), B, 0, stream>>>(ei, h, deg, agg1, E);
    k_relu_bias<<<(int)((totalH + B - 1) / B), B, 0, stream>>>(agg1, b1, totalH);

    // layer 2
    k_gemm2<<<(int)(((long long)N * 32 + B - 1) / B), B, 0, stream>>>(agg1, W2, h2, N);
    k_init_out<<<(N + B - 1) / B, B, 0, stream>>>(h2, deg, b2, out, N);
    k_scatter2<<<(int)((E + B - 1) / B), B, 0, stream>>>(ei, h2, deg, out, E);
}